// DCN_49460843381606
// MI455X (gfx1250) — compile-verified
//
#include <hip/hip_runtime.h>
#include <hip/hip_bf16.h>
#include <math.h>

// ---------------------------------------------------------------------------
// DCN clustering loss on MI455X (gfx1250, wave32).
// Matmuls via V_WMMA_F32_16X16X4_F32 (full fp32 matrix path).
// adj (268 MB) streams once per adj-GEMM (split-K -> 4096 waves in flight)
// and once in the BCE kernel; all other operands are L2-resident (192 MB L2).
// GEMM inner loop: load phase (distinct regs for MT A-frags + 2*NT B-frags)
// then MT*NT WMMAs -> single-counter partial loadcnt waits and scheduler
// overlap of next-step loads with current-step WMMAs. A/B pointers are plain
// scalars so address-space inference keeps every load on the GLOBAL path
// (no flat_load -> no DScnt coupling, no LDS-path waste). OOB handling is
// branchless (clamped address + select) so EXEC is all-ones at every v_wmma.
// ---------------------------------------------------------------------------

typedef float v2f __attribute__((ext_vector_type(2)));
typedef float v8f __attribute__((ext_vector_type(8)));

#define NN      8192
#define BETA    1.0f
#define LAMDA   1.0f
#define GAMA    0.1f
#define DETA    0.1f
#define NADA    0.1f

// ---------------- workspace layout (float offsets) -------------------------
// acc[0]: mse | acc[1..2]: pcluster | acc[3..8]: S_cc (2x3)
// acc[9..20]: cc (2x6) | acc[21]: distSx | acc[22]: distX | acc[23]: bce
static const size_t OFF_ACC = 0;
static const size_t OFF_LSX = 64;                         // latent_Sx  [N,3]
static const size_t OFF_REC = OFF_LSX + (size_t)NN * 3;   // aerec      [N,5]
static const size_t OFF_XW1 = OFF_REC + (size_t)NN * 5;   // features@W1[N,32]
static const size_t OFF_HID = OFF_XW1 + (size_t)NN * 32;  // hidden     [N,32]
static const size_t OFF_HW2 = OFF_HID + (size_t)NN * 32;  // hidden@W2  [N,6]
static const size_t OFF_LX  = OFF_HW2 + (size_t)NN * 6;   // latent_X   [N,6]
static const size_t OFF_CAT = OFF_LX  + (size_t)NN * 6;   // concat     [N,9]
static const size_t OFF_LOG = OFF_CAT + (size_t)NN * 9;   // logits     [N,2]
static const size_t OFF_CID = OFF_LOG + (size_t)NN * 2;   // cluster_id [N,2]
static const size_t OFF_B1  = OFF_CID + (size_t)NN * 2;   // ping [N,500]
static const size_t OFF_B2  = OFF_B1  + (size_t)NN * 500; // pong [N,500]
static const size_t OFF_A2K = OFF_B2  + (size_t)NN * 500; // wide [N,2000]
// total ~25.4M floats (~102 MB) of d_ws.

// ---------------------------------------------------------------------------
// fp32 WMMA GEMM main loop. FULL=true when the whole NT*16 column tile is in
// bounds (no selects). Load phase fills distinct registers, then MT*NT WMMAs.
// A pointers are scalar parameters (MT<=2) so they stay global-address-space.
// ---------------------------------------------------------------------------
template<int MT, int NT, bool FULL>
__device__ __forceinline__ void kloop(const float* Ap0, const float* Ap1,
                                      const float* Bp0, const float* Bp1,
                                      int nsteps, int strideB,
                                      const bool (&colok)[NT],
                                      v8f (&acc)[MT][NT]) {
#pragma unroll 2
  for (int s = 0; s < nsteps; ++s) {
    // ---- load phase: all fragments of this k-step into distinct regs ----
    v2f a[MT];
    a[0].x = Ap0[0];
    a[0].y = Ap0[1];
    Ap0 += 4;
    if (MT > 1) {
      a[MT - 1].x = Ap1[0];
      a[MT - 1].y = Ap1[1];
      Ap1 += 4;
    }
    float bx[NT], by[NT];
#pragma unroll
    for (int t = 0; t < NT; ++t) {
      bx[t] = Bp0[t * 16];        // immediate offsets 0,64,128,192 bytes
      by[t] = Bp1[t * 16];
    }
    Bp0 += strideB;
    Bp1 += strideB;
    // ---- compute phase: MT*NT WMMAs back-to-back ----
#pragma unroll
    for (int t = 0; t < NT; ++t) {
      v2f b;
      b.x = FULL ? bx[t] : (colok[t] ? bx[t] : 0.f);
      b.y = FULL ? by[t] : (colok[t] ? by[t] : 0.f);
#pragma unroll
      for (int i = 0; i < MT; ++i)
        acc[i][t] = __builtin_amdgcn_wmma_f32_16x16x4_f32(
            false, a[i], false, b, (short)0, acc[i][t], false, false);
    }
  }
}

// C[M,N] = act(A[M,K_range] @ B[K_range,N] + bias). One wave = (MT*16) x
// (NT*16) tile. M multiple of MT*64. Split-K via blockIdx.z * kchunk; ATOMIC
// path does global_atomic_add_f32 stores (no bias/relu).
template<int MT, int NT, bool BIAS, bool RELU, bool ATOMIC>
__global__ __launch_bounds__(128)
void gemm_wmma_f32(const float* __restrict__ A, const float* __restrict__ B,
                   const float* __restrict__ bias, float* __restrict__ C,
                   int M, int N, int K, int kchunk) {
  const int lane = threadIdx.x;                 // 0..31
  const int m0   = (blockIdx.y * 4 + threadIdx.y) * (16 * MT);
  const int n0   = blockIdx.x * (NT * 16);
  const int r15  = lane & 15;
  const int hi   = lane >> 4;                   // 0/1: lane half picks K{0,1}/{2,3}
  const int koff = hi * 2;

  const int kb = blockIdx.z * kchunk;
  const int ke = min(K, kb + kchunk);

  int  col[NT];
  int  colc[NT];                                // clamped, always-legal
  bool colok[NT];
#pragma unroll
  for (int t = 0; t < NT; ++t) {
    col[t]   = n0 + t * 16 + r15;
    colok[t] = col[t] < N;
    colc[t]  = colok[t] ? col[t] : (N - 1);
  }
  const bool fullN = (n0 + NT * 16) <= N;       // uniform across the wave

  v8f acc[MT][NT];
#pragma unroll
  for (int i = 0; i < MT; ++i)
#pragma unroll
    for (int t = 0; t < NT; ++t)
      acc[i][t] = (v8f){0.f,0.f,0.f,0.f,0.f,0.f,0.f,0.f};

  // Scalar pointers only (keeps global address space through inference).
  const float* Arow0 = A + (size_t)(m0 + r15) * (size_t)K;
  const float* Arow1 = (MT > 1) ? (A + (size_t)(m0 + 16 + r15) * (size_t)K)
                                : Arow0;
  const float* Ap0 = Arow0 + kb + koff;
  const float* Ap1 = Arow1 + kb + koff;
  // Bp0/Bp1 anchored at clamped column of tile 0; tiles t use +16*t immediates.
  const float* Bp0 = B + (size_t)(kb + koff) * (size_t)N + colc[0];
  const float* Bp1 = Bp0 + N;
  const int strideB = 4 * N;

  const int span  = ke - kb;
  const int steps = span >> 2;                  // full K-steps of 4
  if (fullN) kloop<MT, NT, true >(Ap0, Ap1, Bp0, Bp1, steps, strideB, colok, acc);
  else       kloop<MT, NT, false>(Ap0, Ap1, Bp0, Bp1, steps, strideB, colok, acc);

  if (span & 3) {                               // uniform tail, clamped+selected
    const int k0  = kb + (steps << 2);
    const int ka0 = min(k0 + koff,     ke - 1);
    const int ka1 = min(k0 + koff + 1, ke - 1);
    const bool ok0 = (k0 + koff)     < ke;
    const bool ok1 = (k0 + koff + 1) < ke;
    v2f a[MT];
    {
      float a0 = Arow0[ka0], a1 = Arow0[ka1];
      a[0].x = ok0 ? a0 : 0.f;
      a[0].y = ok1 ? a1 : 0.f;
    }
    if (MT > 1) {
      float a0 = Arow1[ka0], a1 = Arow1[ka1];
      a[MT - 1].x = ok0 ? a0 : 0.f;
      a[MT - 1].y = ok1 ? a1 : 0.f;
    }
    const float* B0 = B + (size_t)ka0 * (size_t)N;
    const float* B1 = B + (size_t)ka1 * (size_t)N;
#pragma unroll
    for (int t = 0; t < NT; ++t) {
      float bx = B0[colc[t]];
      float by = B1[colc[t]];
      v2f b;
      b.x = (ok0 && colok[t]) ? bx : 0.f;
      b.y = (ok1 && colok[t]) ? by : 0.f;
#pragma unroll
      for (int i = 0; i < MT; ++i)
        acc[i][t] = __builtin_amdgcn_wmma_f32_16x16x4_f32(
            false, a[i], false, b, (short)0, acc[i][t], false, false);
    }
  }

  // D layout: acc[i][t][r] -> row m0 + i*16 + r + 8*hi, col col[t]
#pragma unroll
  for (int i = 0; i < MT; ++i)
#pragma unroll
    for (int t = 0; t < NT; ++t) {
      if (!colok[t]) continue;
      float bv = BIAS ? bias[col[t]] : 0.f;
#pragma unroll
      for (int r = 0; r < 8; ++r) {
        float v = acc[i][t][r] + bv;
        if (RELU) v = fmaxf(v, 0.f);
        size_t idx = (size_t)(m0 + i * 16 + r + hi * 8) * (size_t)N + col[t];
        if (ATOMIC) atomicAdd(&C[idx], v);
        else        C[idx] = v;
      }
    }
}

// ---------------------------- small kernels --------------------------------
__global__ void zero_buf_kernel(float* __restrict__ p, int ntot) {
  int i = blockIdx.x * blockDim.x + threadIdx.x;
  if (i < ntot) p[i] = 0.f;
}

__global__ void relu_buf_kernel(float* __restrict__ p, int ntot) {
  int i = blockIdx.x * blockDim.x + threadIdx.x;
  if (i < ntot) p[i] = fmaxf(p[i], 0.f);
}

__global__ void concat_kernel(const float* __restrict__ ls,
                              const float* __restrict__ lx,
                              float* __restrict__ cat, int n) {
  int i = blockIdx.x * blockDim.x + threadIdx.x;
  if (i < n) {
    float* o = cat + (size_t)i * 9;
    const float* a = ls + (size_t)i * 3;
    const float* b = lx + (size_t)i * 6;
    o[0] = a[0]; o[1] = a[1]; o[2] = a[2];
#pragma unroll
    for (int d = 0; d < 6; ++d) o[3 + d] = b[d];
  }
}

__global__ void softmax2_kernel(const float* __restrict__ logits,
                                float* __restrict__ cid, int n) {
  int i = blockIdx.x * blockDim.x + threadIdx.x;
  if (i < n) {
    float a = logits[2 * i], b = logits[2 * i + 1];
    float m = fmaxf(a, b);
    float ea = __expf(a - m), eb = __expf(b - m);
    float s = ea + eb;
    cid[2 * i]     = ea / s;
    cid[2 * i + 1] = eb / s;
  }
}

__device__ __forceinline__ float wred(float v) {
#pragma unroll
  for (int o = 16; o > 0; o >>= 1) v += __shfl_down(v, o, 32);
  return v;   // valid in lane 0 of each wave
}

__global__ void reduce_mse_kernel(const float* __restrict__ Sx,
                                  const float* __restrict__ rec,
                                  float* __restrict__ acc, int total) {
  float s = 0.f;
  for (int i = blockIdx.x * blockDim.x + threadIdx.x; i < total;
       i += gridDim.x * blockDim.x) {
    float d = Sx[i] - rec[i];
    s += d * d;
  }
  s = wred(s);
  if ((threadIdx.x & 31) == 0) atomicAdd(&acc[0], s);
}

__global__ void reduce_centers_kernel(const float* __restrict__ cid,
                                      const float* __restrict__ ls,
                                      const float* __restrict__ lx,
                                      float* __restrict__ acc, int n) {
  float p[2] = {0.f, 0.f};
  float sc[2][3] = {};
  float cc[2][6] = {};
  for (int i = blockIdx.x * blockDim.x + threadIdx.x; i < n;
       i += gridDim.x * blockDim.x) {
    float w0 = cid[2 * i], w1 = cid[2 * i + 1];
    p[0] += w0; p[1] += w1;
#pragma unroll
    for (int d = 0; d < 3; ++d) {
      float v = ls[(size_t)3 * i + d];
      sc[0][d] += w0 * v; sc[1][d] += w1 * v;
    }
#pragma unroll
    for (int d = 0; d < 6; ++d) {
      float v = lx[(size_t)6 * i + d];
      cc[0][d] += w0 * v; cc[1][d] += w1 * v;
    }
  }
  int lane = threadIdx.x & 31;
  float v;
  v = wred(p[0]); if (lane == 0) atomicAdd(&acc[1], v);
  v = wred(p[1]); if (lane == 0) atomicAdd(&acc[2], v);
#pragma unroll
  for (int k = 0; k < 2; ++k)
#pragma unroll
    for (int d = 0; d < 3; ++d) {
      v = wred(sc[k][d]); if (lane == 0) atomicAdd(&acc[3 + k * 3 + d], v);
    }
#pragma unroll
  for (int k = 0; k < 2; ++k)
#pragma unroll
    for (int d = 0; d < 6; ++d) {
      v = wred(cc[k][d]); if (lane == 0) atomicAdd(&acc[9 + k * 6 + d], v);
    }
}

__global__ void reduce_dist_kernel(const float* __restrict__ cid,
                                   const float* __restrict__ ls,
                                   const float* __restrict__ lx,
                                   float* __restrict__ acc, int n) {
  __shared__ float scc[6], ccc[12];
  if (threadIdx.x < 6)  scc[threadIdx.x] = acc[3 + threadIdx.x];
  if (threadIdx.x < 12) ccc[threadIdx.x] = acc[9 + threadIdx.x];
  __syncthreads();
  float dsx = 0.f, dxx = 0.f;
  for (int i = blockIdx.x * blockDim.x + threadIdx.x; i < n;
       i += gridDim.x * blockDim.x) {
    float w[2] = {cid[2 * i], cid[2 * i + 1]};
#pragma unroll
    for (int k = 0; k < 2; ++k) {
      float s = 0.f;
#pragma unroll
      for (int d = 0; d < 3; ++d) {
        float dd = ls[(size_t)3 * i + d] - scc[k * 3 + d];
        s += dd * dd;
      }
      dsx += w[k] * s;
      float s2 = 0.f;
#pragma unroll
      for (int d = 0; d < 6; ++d) {
        float dd = lx[(size_t)6 * i + d] - ccc[k * 6 + d];
        s2 += dd * dd;
      }
      dxx += w[k] * s2;
    }
  }
  int lane = threadIdx.x & 31;
  float v = wred(dsx); if (lane == 0) atomicAdd(&acc[21], v);
  v = wred(dxx);       if (lane == 0) atomicAdd(&acc[22], v);
}

// One block per row i; threads stride columns j. adj row streams coalesced
// (read once total); latent_X (196 KB) stays L2-resident.
__global__ __launch_bounds__(256)
void reduce_bce_kernel(const float* __restrict__ adj,
                       const float* __restrict__ lx,
                       float* __restrict__ acc, int n) {
  const int i = blockIdx.x;
  float zi[6];
#pragma unroll
  for (int d = 0; d < 6; ++d) zi[d] = lx[(size_t)6 * i + d];
  const float* __restrict__ arow = adj + (size_t)i * (size_t)n;
  float s = 0.f;
  for (int j = threadIdx.x; j < n; j += blockDim.x) {
    const float* __restrict__ zj = lx + (size_t)6 * j;
    float z = 0.f;
#pragma unroll
    for (int d = 0; d < 6; ++d) z += zi[d] * zj[d];
    float p = 1.f / (1.f + __expf(-z));
    p = fminf(fmaxf(p, 1e-7f), 1.f - 1e-7f);
    float a = arow[j];
    s += a * __logf(p) + (1.f - a) * log1pf(-p);
  }
  s = wred(s);
  __shared__ float sm[8];
  int lane = threadIdx.x & 31, wid = threadIdx.x >> 5;
  if (lane == 0) sm[wid] = s;
  __syncthreads();
  if (threadIdx.x == 0) {
    float t = 0.f;
    for (int w = 0; w < (int)(blockDim.x >> 5); ++w) t += sm[w];
    atomicAdd(&acc[23], t);
  }
}

__global__ void combine_kernel(const float* __restrict__ acc,
                               float* __restrict__ out, int n) {
  if (threadIdx.x == 0 && blockIdx.x == 0) {
    float aerec_loss = LAMDA * acc[0] / ((float)n * 5.f);
    float bce = -acc[23] / ((float)n * (float)n);
    float gaerec_loss = BETA * 0.5f * bce;
    float clst = 0.f;
    for (int k = 0; k < 2; ++k) {
      float p = acc[1 + k] / (float)n;
      clst += p * __logf(p + 1e-10f);
    }
    clst *= NADA;
    float distSx_loss = 0.5f * DETA * acc[21] / (float)n;
    float dist_loss   = 0.5f * GAMA * acc[22] / (float)n;
    out[0] = aerec_loss + dist_loss + gaerec_loss + distSx_loss + clst;
  }
}

// ------------------------------ host side ----------------------------------
#define GEMM_DISPATCH(MT, NT, GRD)                                                   \
  do {                                                                               \
    if (atomic) {                                                                    \
      gemm_wmma_f32<MT, NT, false, false, true ><<<GRD, blk, 0, s>>>(A,B,bias,C,M,N,K,kchunk); \
    } else if (bias) {                                                               \
      if (relu) gemm_wmma_f32<MT, NT, true,  true,  false><<<GRD, blk, 0, s>>>(A,B,bias,C,M,N,K,kchunk); \
      else      gemm_wmma_f32<MT, NT, true,  false, false><<<GRD, blk, 0, s>>>(A,B,bias,C,M,N,K,kchunk); \
    } else {                                                                         \
      if (relu) gemm_wmma_f32<MT, NT, false, true,  false><<<GRD, blk, 0, s>>>(A,B,bias,C,M,N,K,kchunk); \
      else      gemm_wmma_f32<MT, NT, false, false, false><<<GRD, blk, 0, s>>>(A,B,bias,C,M,N,K,kchunk); \
    }                                                                                \
  } while (0)

static void gemm(const float* A, const float* B, const float* bias, float* C,
                 int M, int N, int K, bool relu, bool atomic, int splitk,
                 hipStream_t s) {
  dim3 blk(32, 4, 1);             // 4 waves/block, one (MT*16)x(NT*16) tile each
  int kchunk = (K + splitk - 1) / splitk;
  if (N > 32) {                    // wide, L2-resident layers: 32x64 per wave
    dim3 grd((unsigned)((N + 63) / 64), (unsigned)(M / 128), (unsigned)splitk);
    GEMM_DISPATCH(2, 4, grd);
  } else if (N > 16) {             // adj @ XW1 (N=32): 16x32 per wave
    dim3 grd((unsigned)((N + 31) / 32), (unsigned)(M / 64), (unsigned)splitk);
    GEMM_DISPATCH(1, 2, grd);
  } else {                         // tiny-N layers: 16x16 per wave
    dim3 grd((unsigned)((N + 15) / 16), (unsigned)(M / 64), (unsigned)splitk);
    GEMM_DISPATCH(1, 1, grd);
  }
}

extern "C" void kernel_launch(void* const* d_in, const int* in_sizes, int n_in,
                              void* d_out, int out_size, void* d_ws, size_t ws_size,
                              hipStream_t stream) {
  (void)in_sizes; (void)n_in; (void)out_size; (void)ws_size;
  const int n = NN;

  // Inputs flattened in setup_inputs() insertion order:
  // Sx, features, adj, enc(W,b)x4, dec(W,b)x4, cl(W,b)x3, gcn W1, gcn W2
  const float* Sx       = (const float*)d_in[0];
  const float* features = (const float*)d_in[1];
  const float* adj      = (const float*)d_in[2];
  const float* eW0 = (const float*)d_in[3];  const float* eB0 = (const float*)d_in[4];
  const float* eW1 = (const float*)d_in[5];  const float* eB1 = (const float*)d_in[6];
  const float* eW2 = (const float*)d_in[7];  const float* eB2 = (const float*)d_in[8];
  const float* eW3 = (const float*)d_in[9];  const float* eB3 = (const float*)d_in[10];
  const float* dW0 = (const float*)d_in[11]; const float* dB0 = (const float*)d_in[12];
  const float* dW1 = (const float*)d_in[13]; const float* dB1 = (const float*)d_in[14];
  const float* dW2 = (const float*)d_in[15]; const float* dB2 = (const float*)d_in[16];
  const float* dW3 = (const float*)d_in[17]; const float* dB3 = (const float*)d_in[18];
  const float* cW0 = (const float*)d_in[19]; const float* cB0 = (const float*)d_in[20];
  const float* cW1 = (const float*)d_in[21]; const float* cB1 = (const float*)d_in[22];
  const float* cW2 = (const float*)d_in[23]; const float* cB2 = (const float*)d_in[24];
  const float* gW1 = (const float*)d_in[25];
  const float* gW2 = (const float*)d_in[26];

  float* ws   = (float*)d_ws;
  float* acc  = ws + OFF_ACC;
  float* lsx  = ws + OFF_LSX;
  float* rec  = ws + OFF_REC;
  float* xw1  = ws + OFF_XW1;
  float* hid  = ws + OFF_HID;
  float* hw2  = ws + OFF_HW2;
  float* lx   = ws + OFF_LX;
  float* cat  = ws + OFF_CAT;
  float* logi = ws + OFF_LOG;
  float* cid  = ws + OFF_CID;
  float* b1   = ws + OFF_B1;
  float* b2   = ws + OFF_B2;
  float* a2k  = ws + OFF_A2K;

  zero_buf_kernel<<<1, 64, 0, stream>>>(acc, 64);

  // --- AutoEncoder ---
  gemm(Sx,  eW0, eB0, b1,  n, 500,  5,    true,  false, 1, stream);
  gemm(b1,  eW1, eB1, b2,  n, 500,  500,  true,  false, 1, stream);
  gemm(b2,  eW2, eB2, a2k, n, 2000, 500,  true,  false, 1, stream);
  gemm(a2k, eW3, eB3, lsx, n, 3,    2000, false, false, 1, stream);   // latent_Sx
  gemm(lsx, dW0, dB0, a2k, n, 2000, 3,    true,  false, 1, stream);
  gemm(a2k, dW1, dB1, b1,  n, 500,  2000, true,  false, 1, stream);
  gemm(b1,  dW2, dB2, b2,  n, 500,  500,  true,  false, 1, stream);
  gemm(b2,  dW3, dB3, rec, n, 5,    500,  false, false, 1, stream);   // aerec

  // --- GCN: hidden = relu(adj @ (features@W1)); latent_X = adj @ (hidden@W2)
  gemm(features, gW1, nullptr, xw1, n, 32, 128, false, false, 1, stream);
  zero_buf_kernel<<<(n * 32 + 255) / 256, 256, 0, stream>>>(hid, n * 32);
  gemm(adj, xw1, nullptr, hid, n, 32, n, false, true, 8, stream);  // split-K 8
  relu_buf_kernel<<<(n * 32 + 255) / 256, 256, 0, stream>>>(hid, n * 32);
  gemm(hid, gW2, nullptr, hw2, n, 6, 32, false, false, 1, stream);
  zero_buf_kernel<<<(n * 6 + 255) / 256, 256, 0, stream>>>(lx, n * 6);
  gemm(adj, hw2, nullptr, lx, n, 6, n, false, true, 8, stream);    // split-K 8

  // --- Cluster head ---
  concat_kernel<<<(n + 255) / 256, 256, 0, stream>>>(lsx, lx, cat, n);
  gemm(cat, cW0, cB0, b1,   n, 500, 9,   true,  false, 1, stream);
  gemm(b1,  cW1, cB1, b2,   n, 500, 500, true,  false, 1, stream);
  gemm(b2,  cW2, cB2, logi, n, 2,   500, false, false, 1, stream);
  softmax2_kernel<<<(n + 255) / 256, 256, 0, stream>>>(logi, cid, n);

  // --- Reductions ---
  reduce_mse_kernel<<<64, 256, 0, stream>>>(Sx, rec, acc, n * 5);
  reduce_centers_kernel<<<64, 256, 0, stream>>>(cid, lsx, lx, acc, n);
  reduce_dist_kernel<<<64, 256, 0, stream>>>(cid, lsx, lx, acc, n);
  reduce_bce_kernel<<<n, 256, 0, stream>>>(adj, lx, acc, n);

  combine_kernel<<<1, 32, 0, stream>>>(acc, (float*)d_out, n);
}